// MyFirstGNN_32332513804719
// MI455X (gfx1250) — compile-verified
//
#include <hip/hip_runtime.h>
#include <cstdint>
#include <cstddef>

#define N_NODES 50000
#define N_EDGES 800000
#define FDIM    128
#define N_LABELS 16
#define ALPHA   0.2f

typedef __attribute__((ext_vector_type(2))) float v2f;
typedef __attribute__((ext_vector_type(8))) float v8f;

// ---------------------------------------------------------------------------
// f32 WMMA: D(16x16) = A(16x4) * B(4x16) + C   (V_WMMA_F32_16X16X4_F32)
// ---------------------------------------------------------------------------
__device__ __forceinline__ v8f wmma_f32_16x16x4(v2f a, v2f b, v8f c) {
    return __builtin_amdgcn_wmma_f32_16x16x4_f32(
        /*neg_a=*/false, a, /*neg_b=*/false, b,
        /*c_mod=*/(short)0, c, /*reuse_a=*/false, /*reuse_b=*/false);
}

__device__ __forceinline__ void atomAddF32(float* p, float v) {
    (void)__hip_atomic_fetch_add(p, v, __ATOMIC_RELAXED, __HIP_MEMORY_SCOPE_AGENT);
}

// ---------------------------------------------------------------------------
// Degree histograms: deg_out[src]++, deg_in[dst]++ (float atomics)
// ---------------------------------------------------------------------------
__global__ void deg_kernel(const int* __restrict__ src, const int* __restrict__ dst,
                           float* __restrict__ deg_out, float* __restrict__ deg_in) {
    int e = blockIdx.x * blockDim.x + threadIdx.x;
    if (e >= N_EDGES) return;
    atomAddF32(&deg_out[src[e]], 1.0f);
    atomAddF32(&deg_in[dst[e]], 1.0f);
}

// ---------------------------------------------------------------------------
// Per-edge symmetric normalization: rsqrt(max(deg_out[s],1)*max(deg_in[d],1))
// ---------------------------------------------------------------------------
__global__ void norm_kernel(const int* __restrict__ src, const int* __restrict__ dst,
                            const float* __restrict__ deg_out, const float* __restrict__ deg_in,
                            float* __restrict__ norm) {
    int e = blockIdx.x * blockDim.x + threadIdx.x;
    if (e >= N_EDGES) return;
    float po = fmaxf(deg_out[src[e]], 1.0f);
    float pi = fmaxf(deg_in[dst[e]], 1.0f);
    norm[e] = rsqrtf(po * pi);
}

// ---------------------------------------------------------------------------
// Aggregation: agg[dst] += h[src] * norm.  One wave32 per edge; each lane
// owns a float4 slice (32 lanes * 4 = 128 features). Gathers are 512B
// coalesced; scatter uses non-returning global_atomic_add_f32 (L2-resident).
// ---------------------------------------------------------------------------
__global__ void agg_kernel(const float* __restrict__ h, const int* __restrict__ src,
                           const int* __restrict__ dst, const float* __restrict__ norm,
                           float* __restrict__ agg) {
    int gid  = blockIdx.x * blockDim.x + threadIdx.x;
    int e    = gid >> 5;
    int lane = gid & 31;
    if (e >= N_EDGES) return;
    int   s  = src[e];
    int   d  = dst[e];
    float nv = norm[e];
    float4 v = ((const float4*)(h + (size_t)s * FDIM))[lane];
    float* op = agg + (size_t)d * FDIM + lane * 4;
    atomAddF32(op + 0, v.x * nv);
    atomAddF32(op + 1, v.y * nv);
    atomAddF32(op + 2, v.z * nv);
    atomAddF32(op + 3, v.w * nv);
}

// ---------------------------------------------------------------------------
// GCSConv matmul: out = leaky(agg @ W1 + h @ W2 + b), all [*,128]@[128,128].
// One wave per 16x16 tile; blockIdx.x -> M tile (50000/16 = 3125 exact),
// wave-in-block (8 waves) -> N tile (128/16 = 8). K-loop: 32 x 16x16x4 WMMA
// per weight matrix, both products chained through one accumulator.
// ---------------------------------------------------------------------------
__global__ void __launch_bounds__(256)
gcs_kernel(const float* __restrict__ agg, const float* __restrict__ h,
           const float* __restrict__ w1, const float* __restrict__ w2,
           const float* __restrict__ bias, float* __restrict__ out) {
    const int lane = threadIdx.x & 31;
    const int wv   = threadIdx.x >> 5;      // 0..7 -> N tile
    const int m0   = blockIdx.x * 16;
    const int n0   = wv * 16;
    const int l15  = lane & 15;
    const int hi   = lane >> 4;             // 0 or 1 (half-wave)
    const int arow = m0 + l15;              // A: row per lane
    const int bcol = n0 + l15;              // B: column per lane

    v8f acc = {0.f, 0.f, 0.f, 0.f, 0.f, 0.f, 0.f, 0.f};

    for (int k = 0; k < FDIM; k += 4) {
        const int kk = k + (hi << 1);       // K=0,1 (lanes 0-15) / K=2,3 (lanes 16-31)
        v2f aA = *(const v2f*)(agg + (size_t)arow * FDIM + kk);
        v2f aH = *(const v2f*)(h   + (size_t)arow * FDIM + kk);
        v2f b1 = { w1[(size_t)kk * FDIM + bcol], w1[(size_t)(kk + 1) * FDIM + bcol] };
        v2f b2 = { w2[(size_t)kk * FDIM + bcol], w2[(size_t)(kk + 1) * FDIM + bcol] };
        acc = wmma_f32_16x16x4(aA, b1, acc);
        acc = wmma_f32_16x16x4(aH, b2, acc);
    }

    // C/D layout: VGPR r, lanes 0-15 -> (m0+r, n0+lane); lanes 16-31 -> (m0+8+r, ...)
    const float bb = bias[bcol];
    float* op = out + (size_t)(m0 + (hi << 3)) * FDIM + bcol;
#pragma unroll
    for (int r = 0; r < 8; ++r) {
        float v = acc[r] + bb;
        op[(size_t)r * FDIM] = (v >= 0.0f) ? v : ALPHA * v;
    }
}

// ---------------------------------------------------------------------------
// Dense head: out = tanh(h @ W + b), [*,128]@[128,16]. One wave per 16x16
// tile (single N tile). Guard at wave granularity -> EXEC all-1s for WMMA.
// ---------------------------------------------------------------------------
__global__ void __launch_bounds__(256)
dense_kernel(const float* __restrict__ h, const float* __restrict__ w,
             const float* __restrict__ bias, float* __restrict__ out) {
    const int lane = threadIdx.x & 31;
    const int wv   = threadIdx.x >> 5;
    const int m0   = (blockIdx.x * 8 + wv) * 16;
    if (m0 >= N_NODES) return;
    const int l15  = lane & 15;
    const int hi   = lane >> 4;
    const int arow = m0 + l15;

    v8f acc = {0.f, 0.f, 0.f, 0.f, 0.f, 0.f, 0.f, 0.f};

    for (int k = 0; k < FDIM; k += 4) {
        const int kk = k + (hi << 1);
        v2f a = *(const v2f*)(h + (size_t)arow * FDIM + kk);
        v2f b = { w[(size_t)kk * N_LABELS + l15], w[(size_t)(kk + 1) * N_LABELS + l15] };
        acc = wmma_f32_16x16x4(a, b, acc);
    }

    const float bb = bias[l15];
    float* op = out + (size_t)(m0 + (hi << 3)) * N_LABELS + l15;
#pragma unroll
    for (int r = 0; r < 8; ++r)
        op[(size_t)r * N_LABELS] = tanhf(acc[r] + bb);
}

// ---------------------------------------------------------------------------
// Launch
// ---------------------------------------------------------------------------
extern "C" void kernel_launch(void* const* d_in, const int* in_sizes, int n_in,
                              void* d_out, int out_size, void* d_ws, size_t ws_size,
                              hipStream_t stream) {
    (void)in_sizes; (void)n_in; (void)out_size; (void)ws_size;

    const float* x     = (const float*)d_in[0];
    const int*   ei    = (const int*)  d_in[1];
    const int*   src   = ei;                 // edge_index[0]
    const int*   dst   = ei + N_EDGES;       // edge_index[1]
    const float* w1_1  = (const float*)d_in[2];
    const float* w2_1  = (const float*)d_in[3];
    const float* b_1   = (const float*)d_in[4];
    const float* w1_2  = (const float*)d_in[5];
    const float* w2_2  = (const float*)d_in[6];
    const float* b_2   = (const float*)d_in[7];
    const float* w1_3  = (const float*)d_in[8];
    const float* w2_3  = (const float*)d_in[9];
    const float* b_3   = (const float*)d_in[10];
    const float* w_d   = (const float*)d_in[11];
    const float* b_d   = (const float*)d_in[12];
    float*       out   = (float*)d_out;

    // Workspace layout (256B aligned slices)
    const size_t FEAT_BYTES = (size_t)N_NODES * FDIM * sizeof(float);   // 25.6 MB
    size_t off = 0;
    auto alloc = [&](size_t bytes) {
        void* p = (char*)d_ws + off;
        off += (bytes + 255) & ~(size_t)255;
        return p;
    };
    float* norm    = (float*)alloc((size_t)N_EDGES * sizeof(float));
    float* deg_out = (float*)alloc((size_t)N_NODES * sizeof(float));
    float* deg_in  = (float*)alloc((size_t)N_NODES * sizeof(float));
    float* agg     = (float*)alloc(FEAT_BYTES);
    float* hA      = (float*)alloc(FEAT_BYTES);
    float* hB      = (float*)alloc(FEAT_BYTES);

    const int EDGE_BLOCKS = (N_EDGES + 255) / 256;        // 1 thread / edge
    const int AGG_BLOCKS  = (N_EDGES * 32) / 256;         // 1 wave32 / edge
    const int MM_BLOCKS   = N_NODES / 16;                 // 3125, exact
    const int DENSE_BLOCKS = (N_NODES / 16 + 7) / 8;      // 391

    // Edge normalization
    hipMemsetAsync(deg_out, 0, (size_t)N_NODES * sizeof(float), stream);
    hipMemsetAsync(deg_in,  0, (size_t)N_NODES * sizeof(float), stream);
    deg_kernel <<<EDGE_BLOCKS, 256, 0, stream>>>(src, dst, deg_out, deg_in);
    norm_kernel<<<EDGE_BLOCKS, 256, 0, stream>>>(src, dst, deg_out, deg_in, norm);

    // Layer 1: x -> hA
    hipMemsetAsync(agg, 0, FEAT_BYTES, stream);
    agg_kernel<<<AGG_BLOCKS, 256, 0, stream>>>(x, src, dst, norm, agg);
    gcs_kernel<<<MM_BLOCKS, 256, 0, stream>>>(agg, x, w1_1, w2_1, b_1, hA);

    // Layer 2: hA -> hB
    hipMemsetAsync(agg, 0, FEAT_BYTES, stream);
    agg_kernel<<<AGG_BLOCKS, 256, 0, stream>>>(hA, src, dst, norm, agg);
    gcs_kernel<<<MM_BLOCKS, 256, 0, stream>>>(agg, hA, w1_2, w2_2, b_2, hB);

    // Layer 3: hB -> hA
    hipMemsetAsync(agg, 0, FEAT_BYTES, stream);
    agg_kernel<<<AGG_BLOCKS, 256, 0, stream>>>(hB, src, dst, norm, agg);
    gcs_kernel<<<MM_BLOCKS, 256, 0, stream>>>(agg, hB, w1_3, w2_3, b_3, hA);

    // Dense head: hA -> out
    dense_kernel<<<DENSE_BLOCKS, 256, 0, stream>>>(hA, w_d, b_d, out);
}